// Unfold_10780367913135
// MI455X (gfx1250) — compile-verified
//
#include <hip/hip_runtime.h>

typedef float v4f __attribute__((ext_vector_type(4)));
typedef int   v4i __attribute__((ext_vector_type(4)));
typedef __attribute__((address_space(1))) v4i gas_v4i;   // global (AS1)
typedef __attribute__((address_space(3))) v4i las_v4i;   // LDS (AS3)

#define H_DIM 56
#define W_DIM 56
#define HW 3136            // 56*56
#define HW4 784            // float4s per plane
#define LDS_STRIDE 64
#define LDS_ROWS 59        // 1 guard row + 58 padded rows

#if __has_builtin(__builtin_amdgcn_global_load_async_to_lds_b128)
#define UNFOLD_ASYNC 1
#else
#define UNFOLD_ASYNC 0
#endif

__global__ __launch_bounds__(256) void unfold3x3_kernel(const float* __restrict__ x,
                                                        float* __restrict__ out)
{
    __shared__ float lds[LDS_ROWS * LDS_STRIDE];   // 15,104 B
    const int tid = threadIdx.x;
    const int bc  = blockIdx.x;                    // b*96 + c, 0..1535
    const float* gplane = x + (long)bc * HW;

    // Kick the HBM fetch of this plane early (gfx1250 global_prefetch_b8)
    // while we zero LDS. 12544 floats = 392 x 128B lines.
    __builtin_prefetch(gplane + tid * 32, 0, 3);
    if (tid < 136) __builtin_prefetch(gplane + (tid + 256) * 32, 0, 3);

    // Phase 1: zero-fill LDS; the untouched borders realize the conv padding.
    v4f zero = {0.f, 0.f, 0.f, 0.f};
    for (int j = tid; j < (LDS_ROWS * LDS_STRIDE) / 4; j += 256)
        ((v4f*)lds)[j] = zero;
    __syncthreads();

    // Phase 2: input row r -> LDS row r+2, cols 0..55 (aligned 128b),
    // via gfx1250 async global->LDS DMA (tracked by ASYNCcnt).
#if UNFOLD_ASYNC
    for (int j = tid; j < HW4; j += 256) {
        int row = j / 14;
        int c4  = j - row * 14;
        __builtin_amdgcn_global_load_async_to_lds_b128(
            (gas_v4i*)(gplane + j * 4),
            (las_v4i*)&lds[(row + 2) * LDS_STRIDE + c4 * 4],
            0, 0);
    }
#if __has_builtin(__builtin_amdgcn_s_wait_asynccnt)
    __builtin_amdgcn_s_wait_asynccnt(0);
#else
    asm volatile("s_wait_asynccnt 0" ::: "memory");
#endif
#else
    for (int j = tid; j < HW4; j += 256) {
        v4f v = ((const v4f*)gplane)[j];
        int row = j / 14;
        int c4  = j - row * 14;
        *(v4f*)&lds[(row + 2) * LDS_STRIDE + c4 * 4] = v;
    }
#endif
    __syncthreads();

    // Phase 3: emit 9 shifted copies. Padded (ph,pw) = lds[(ph+1)*64 + pw - 1].
    // Output is write-once streaming (173 MB): non-temporal 128b stores.
    float* oplane = out + (long)bc * 9 * HW;
    for (int i = tid; i < 9 * HW4; i += 256) {
        int k  = i / HW4;            // 0..8  (ki*3 + kj)
        int r  = i - k * HW4;        // 0..783
        int h  = r / 14;
        int w  = (r - h * 14) * 4;
        int ki = k / 3;
        int kj = k - ki * 3;
        const float* s = &lds[(h + ki + 1) * LDS_STRIDE + (w + kj - 1)];
        v4f v = {s[0], s[1], s[2], s[3]};
        __builtin_nontemporal_store(v, (v4f*)(oplane + k * HW + h * W_DIM + w));
    }
}

extern "C" void kernel_launch(void* const* d_in, const int* in_sizes, int n_in,
                              void* d_out, int out_size, void* d_ws, size_t ws_size,
                              hipStream_t stream)
{
    const float* x = (const float*)d_in[0];
    float* out = (float*)d_out;
    int planes = in_sizes[0] / HW;   // 16*96 = 1536
    unfold3x3_kernel<<<planes, 256, 0, stream>>>(x, out);
    (void)n_in; (void)out_size; (void)d_ws; (void)ws_size;
}